// DCT2_torch_2765958939274
// MI455X (gfx1250) — compile-verified
//
#include <hip/hip_runtime.h>
#include <hip/hip_bf16.h>

typedef __attribute__((ext_vector_type(2))) float v2f;
typedef __attribute__((ext_vector_type(8))) float v8f;

#define N512 512
#define TILE 128          // block tile (M and N)
#define BK   16           // k-slice per LDS stage
#define LDSS 20           // padded LDS row stride (floats): 80B rows, 16B aligned, conflict-spread
#define NKB  (N512 / BK)  // 32 k-iterations

// ---------------------------------------------------------------------------
// Build the orthonormal DCT-II matrix D[k][n] = s_k * cos(pi*(2n+1)*k/(2N))
// s_0 = 1/sqrt(512), s_k = sqrt(2/512) = 1/16 exactly.
// Angle reduced exactly mod 2*pi via integer mod 2048 before cosf.
// ---------------------------------------------------------------------------
__global__ void gen_dct_matrix(float* __restrict__ D) {
    int idx = blockIdx.x * blockDim.x + threadIdx.x;   // 0 .. 512*512-1
    int k = idx >> 9;
    int n = idx & 511;
    int t = ((2 * n + 1) * k) & 2047;                  // period of cos is 2048 in these units
    float ang = (float)t * 3.0679615757712823e-3f;     // pi/1024
    float s = (k == 0) ? 0.04419417382415922f : 0.0625f;
    D[idx] = s * cosf(ang);
}

// ---------------------------------------------------------------------------
// Batched NT GEMM:  C(MxN) = A(MxK, row-major) * B(NxK, row-major)^T
// M = N = K = 512.  blockIdx.z = batch.  strides are element strides per batch
// (0 => shared operand).  TRANS_STORE stores C transposed: C[n*512 + m].
// 128 threads = 4 waves; each wave owns a 64x64 quadrant of the 128x128 tile,
// computed as 4x4 grid of 16x16 WMMA accumulators (v_wmma_f32_16x16x4_f32).
// ---------------------------------------------------------------------------
template <bool TRANS_STORE>
__global__ __launch_bounds__(128)
void gemm_nt_dct(const float* __restrict__ A, long long sA,
                 const float* __restrict__ B, long long sB,
                 float* __restrict__ C, long long sC)
{
    __shared__ __align__(16) float la[2][TILE * LDSS];
    __shared__ __align__(16) float lb[2][TILE * LDSS];

    const int t    = threadIdx.x;
    const int lane = t & 31;
    const int wave = t >> 5;
    const int wr   = wave >> 1;      // wave row   (0..1) -> 64-row band
    const int wc   = wave & 1;       // wave col   (0..1) -> 64-col band
    const int l16  = lane & 15;
    const int khalf = (lane >> 4);   // 0: K pair {0,1}, 1: K pair {2,3}

    const int bM = blockIdx.x * TILE;
    const int bN = blockIdx.y * TILE;

    A += (long long)blockIdx.z * sA + (long long)bM * N512;
    B += (long long)blockIdx.z * sB + (long long)bN * N512;
    C += (long long)blockIdx.z * sC;

    v8f acc[4][4] = {};

    // Each thread owns one row (t) of the A tile and one row of the B tile:
    // 16 floats = 4 float4 each, per k-slice.
    const float4* ga = (const float4*)(A + (long long)t * N512);
    const float4* gb = (const float4*)(B + (long long)t * N512);

    float4 ra[4], rb[4];

    // ---- prologue: load k-slice 0 into buffer 0 ----
#pragma unroll
    for (int i = 0; i < 4; ++i) { ra[i] = ga[i]; rb[i] = gb[i]; }
    {
        float4* da = (float4*)&la[0][t * LDSS];
        float4* db = (float4*)&lb[0][t * LDSS];
#pragma unroll
        for (int i = 0; i < 4; ++i) { da[i] = ra[i]; db[i] = rb[i]; }
    }
    __syncthreads();

    for (int kb = 0; kb < NKB; ++kb) {
        const int cur = kb & 1;

        // prefetch next k-slice from global while computing on current
        if (kb + 1 < NKB) {
            const float4* gan = ga + (kb + 1) * (BK / 4);
            const float4* gbn = gb + (kb + 1) * (BK / 4);
#pragma unroll
            for (int i = 0; i < 4; ++i) { ra[i] = gan[i]; rb[i] = gbn[i]; }
        }

        // ---- compute: 4 k-steps of 4, 16 WMMAs each ----
#pragma unroll
        for (int kk = 0; kk < BK / 4; ++kk) {
            const int kf = kk * 4 + khalf * 2;
            v2f af[4], bf[4];
#pragma unroll
            for (int i = 0; i < 4; ++i)
                af[i] = *(const v2f*)&la[cur][(wr * 64 + i * 16 + l16) * LDSS + kf];
#pragma unroll
            for (int j = 0; j < 4; ++j)
                bf[j] = *(const v2f*)&lb[cur][(wc * 64 + j * 16 + l16) * LDSS + kf];
#pragma unroll
            for (int i = 0; i < 4; ++i)
#pragma unroll
                for (int j = 0; j < 4; ++j)
                    acc[i][j] = __builtin_amdgcn_wmma_f32_16x16x4_f32(
                        false, af[i], false, bf[j], (short)0, acc[i][j],
                        false, false);
        }

        if (kb + 1 < NKB) {
            const int nxt = cur ^ 1;
            float4* da = (float4*)&la[nxt][t * LDSS];
            float4* db = (float4*)&lb[nxt][t * LDSS];
#pragma unroll
            for (int i = 0; i < 4; ++i) { da[i] = ra[i]; db[i] = rb[i]; }
            __syncthreads();
        }
    }

    // ---- epilogue ----
    // C/D layout: acc element r <-> m = tileM + khalf*8 + r, n = tileN + l16.
#pragma unroll
    for (int i = 0; i < 4; ++i) {
#pragma unroll
        for (int j = 0; j < 4; ++j) {
            const int m0 = bM + wr * 64 + i * 16 + khalf * 8;
            const int n  = bN + wc * 64 + j * 16 + l16;
            if (TRANS_STORE) {
                // C[n][m] : 8 consecutive floats per lane -> two float4 stores
                float4 lo = make_float4(acc[i][j][0], acc[i][j][1],
                                        acc[i][j][2], acc[i][j][3]);
                float4 hi = make_float4(acc[i][j][4], acc[i][j][5],
                                        acc[i][j][6], acc[i][j][7]);
                *(float4*)&C[(long long)n * N512 + m0]     = lo;
                *(float4*)&C[(long long)n * N512 + m0 + 4] = hi;
            } else {
#pragma unroll
                for (int r = 0; r < 8; ++r)
                    C[(long long)(m0 + r) * N512 + n] = acc[i][j][r];
            }
        }
    }
}

// ---------------------------------------------------------------------------
// Host launcher.
//   Stage 1: Tt = (X * D^T)^T   (NT gemm, transposed store)  Tt[kw][h]
//   Stage 2: Y  = D * Tt^T      (NT gemm, normal store)      Y[kh][kw]
// Workspace: [ D matrix (1 MB) | Tt chunk buffer ]; images processed in
// chunks sized to fit the available workspace.
// ---------------------------------------------------------------------------
extern "C" void kernel_launch(void* const* d_in, const int* in_sizes, int n_in,
                              void* d_out, int out_size, void* d_ws, size_t ws_size,
                              hipStream_t stream) {
    (void)n_in; (void)out_size;
    const float* x   = (const float*)d_in[0];
    float*       out = (float*)d_out;

    const size_t imgElems = (size_t)N512 * N512;
    const int nImgs = in_sizes[0] / (int)imgElems;   // 8*32 = 256

    float* D = (float*)d_ws;
    float* T = D + imgElems;

    size_t availBytes = (ws_size > imgElems * sizeof(float))
                            ? ws_size - imgElems * sizeof(float) : 0;
    int chunk = (int)(availBytes / (imgElems * sizeof(float)));
    if (chunk < 1) chunk = 1;
    if (chunk > nImgs) chunk = nImgs;

    gen_dct_matrix<<<dim3((N512 * N512) / 256), dim3(256), 0, stream>>>(D);

    for (int off = 0; off < nImgs; off += chunk) {
        const int nb = (off + chunk <= nImgs) ? chunk : (nImgs - off);
        dim3 grid(N512 / TILE, N512 / TILE, nb);   // 4 x 4 x nb
        dim3 blk(128);
        // Stage 1: A = X[img], B = D (shared), C = Tt (transposed store)
        gemm_nt_dct<true><<<grid, blk, 0, stream>>>(
            x + (size_t)off * imgElems, (long long)imgElems,
            D, 0LL,
            T, (long long)imgElems);
        // Stage 2: A = D (shared), B = Tt, C = Y[img]
        gemm_nt_dct<false><<<grid, blk, 0, stream>>>(
            D, 0LL,
            T, (long long)imgElems,
            out + (size_t)off * imgElems, (long long)imgElems);
    }
}